// GraphConvolution_GI_79989470921007
// MI455X (gfx1250) — compile-verified
//
#include <hip/hip_runtime.h>
#include <math.h>

// MI455X / gfx1250, wave32. f32 WMMA 16x16x4 for all dense GEMM stages.

typedef __attribute__((ext_vector_type(2))) float v2f;
typedef __attribute__((ext_vector_type(8))) float v8f;

#define NN 1024   // nodes
#define FD 128    // features = out features

// ---------------------------------------------------------------------------
// Generic 16x16 output tile GEMM step using V_WMMA_F32_16X16X4_F32.
//   A element:  Arow[k]            (row-major A, Arow pre-offset to row i0+m)
//   B element:  Bcol[k * bstride]  (Bcol pre-offset to column g0+n)
// Per ISA 7.12.2 (32-bit operands, wave32):
//   A 16x4 : lane m=lane&15, VGPR v holds K = 2*(lane>>4) + v
//   B 4x16 : lane n=lane&15, VGPR v holds K = 2*(lane>>4) + v
//   C/D    : VGPR r holds row r + 8*(lane>>4), col lane&15
// ---------------------------------------------------------------------------
__device__ __forceinline__ v8f wmma_gemm_f32(const float* __restrict__ Arow,
                                             const float* __restrict__ Bcol,
                                             int bstride, int K, int hi) {
  v8f acc = {};
#pragma unroll 4
  for (int k = 0; k < K; k += 4) {
    const int ka = k + 2 * hi;
    v2f A, B;
    A[0] = Arow[ka];
    A[1] = Arow[ka + 1];
    B[0] = Bcol[(size_t)ka * bstride];
    B[1] = Bcol[(size_t)(ka + 1) * bstride];
    acc = __builtin_amdgcn_wmma_f32_16x16x4_f32(false, A, false, B, (short)0,
                                                acc, false, false);
  }
  return acc;
}

// ---------------------------------------------------------------------------
// Stage 1: a[j,f] = sum_k input[j,k] * w1[f,k]          (blockIdx.y == 0)
//          c[i,f] = sum_k input[i,k] * w1[f,128+k] + b1[f]  (blockIdx.y == 1)
// Tiles: 64 (rows) x 8 (f) = 512 per half; 4 waves/block.
// ---------------------------------------------------------------------------
__global__ void k_ac(const float* __restrict__ input, const float* __restrict__ w1,
                     const float* __restrict__ b1, float* __restrict__ a_out,
                     float* __restrict__ c_out) {
  const int lane = threadIdx.x & 31;
  const int wid  = threadIdx.x >> 5;
  const int tile = blockIdx.x * 4 + wid;      // 0..511
  const int half = blockIdx.y;                // 0 -> a, 1 -> c
  const int j0 = (tile >> 3) * 16;
  const int f0 = (tile & 7) * 16;
  const int m  = lane & 15;
  const int hi = lane >> 4;
  const int off = half * FD;

  const float* Arow = input + (size_t)(j0 + m) * FD;
  const float* Bcol = w1 + (size_t)(f0 + m) * (2 * FD) + off;  // stride 1 in k
  v8f acc = wmma_gemm_f32(Arow, Bcol, 1, FD, hi);

  float* outp = half ? c_out : a_out;
  const float badd = half ? b1[f0 + m] : 0.0f;
#pragma unroll
  for (int r = 0; r < 8; ++r) {
    const int row = r + 8 * hi;
    outp[(size_t)(j0 + row) * FD + f0 + m] = acc[r] + badd;
  }
}

// ---------------------------------------------------------------------------
// Stage 2 (dominant cost, 134M sigmoids):
//   S[i,j] = adj[i,j] * sum_f sigmoid(a[j,f] + c[i,f])
// 32x32 output tile per 256-thread block; both f-panels staged in LDS.
// ---------------------------------------------------------------------------
#define LSTRIDE 136   // pad 128 -> 136 floats: keeps 16B alignment, skews banks

__global__ void k_sig(const float* __restrict__ a, const float* __restrict__ c,
                      const float* __restrict__ adj, float* __restrict__ S) {
  __shared__ float la[32][LSTRIDE];
  __shared__ float lc[32][LSTRIDE];
  const int t  = threadIdx.x;           // 0..255
  const int j0 = blockIdx.x * 32;
  const int i0 = blockIdx.y * 32;

  // Stage tiles: 32 rows x 128 floats each = 1024 float4 per tile, 4/thread.
#pragma unroll
  for (int v = 0; v < 4; ++v) {
    const int idx = t + v * 256;
    const int row = idx >> 5;            // 32 float4 per row
    const int col = (idx & 31) << 2;
    *(float4*)&la[row][col] = *(const float4*)(a + (size_t)(j0 + row) * FD + col);
    *(float4*)&lc[row][col] = *(const float4*)(c + (size_t)(i0 + row) * FD + col);
  }
  __syncthreads();

  const int i  = t >> 3;                 // 0..31
  const int jq = t & 7;                  // 4 consecutive j per thread
  float sums[4] = {0.f, 0.f, 0.f, 0.f};

  for (int f = 0; f < FD; f += 4) {
    const float4 cf = *(const float4*)&lc[i][f];
#pragma unroll
    for (int jj = 0; jj < 4; ++jj) {
      const float4 af = *(const float4*)&la[jq * 4 + jj][f];
      sums[jj] += 1.0f / (1.0f + __expf(-(af.x + cf.x)))
                + 1.0f / (1.0f + __expf(-(af.y + cf.y)))
                + 1.0f / (1.0f + __expf(-(af.z + cf.z)))
                + 1.0f / (1.0f + __expf(-(af.w + cf.w)));
    }
  }

  const size_t o = (size_t)(i0 + i) * NN + j0 + jq * 4;
  const float4 ad = *(const float4*)(adj + o);
  float4 r;
  r.x = ad.x * sums[0];
  r.y = ad.y * sums[1];
  r.z = ad.z * sums[2];
  r.w = ad.w * sums[3];
  *(float4*)(S + o) = r;
}

// ---------------------------------------------------------------------------
// Stage 3: support[i,g] = sum_j S[i,j] * input[j,g]   (K = 1024, WMMA f32)
// ---------------------------------------------------------------------------
__global__ void k_support(const float* __restrict__ S, const float* __restrict__ input,
                          float* __restrict__ support) {
  const int lane = threadIdx.x & 31;
  const int wid  = threadIdx.x >> 5;
  const int tile = blockIdx.x * 4 + wid;   // 0..511
  const int i0 = (tile >> 3) * 16;
  const int g0 = (tile & 7) * 16;
  const int m  = lane & 15;
  const int hi = lane >> 4;

  const float* Arow = S + (size_t)(i0 + m) * NN;
  const float* Bcol = input + g0 + m;       // stride FD in k
  v8f acc = wmma_gemm_f32(Arow, Bcol, FD, NN, hi);

#pragma unroll
  for (int r = 0; r < 8; ++r) {
    const int row = r + 8 * hi;
    support[(size_t)(i0 + row) * FD + g0 + m] = acc[r];
  }
}

// ---------------------------------------------------------------------------
// Stage 4: out = support @ weight + bias   (K = 128, WMMA f32)
// ---------------------------------------------------------------------------
__global__ void k_out(const float* __restrict__ support, const float* __restrict__ weight,
                      const float* __restrict__ bias, float* __restrict__ out) {
  const int lane = threadIdx.x & 31;
  const int wid  = threadIdx.x >> 5;
  const int tile = blockIdx.x * 4 + wid;   // 0..511
  const int i0 = (tile >> 3) * 16;
  const int g0 = (tile & 7) * 16;
  const int m  = lane & 15;
  const int hi = lane >> 4;

  const float* Arow = support + (size_t)(i0 + m) * FD;
  const float* Bcol = weight + g0 + m;      // weight is [F, OUT] row-major
  v8f acc = wmma_gemm_f32(Arow, Bcol, FD, FD, hi);

  const float badd = bias[g0 + m];
#pragma unroll
  for (int r = 0; r < 8; ++r) {
    const int row = r + 8 * hi;
    out[(size_t)(i0 + row) * FD + g0 + m] = acc[r] + badd;
  }
}

// ---------------------------------------------------------------------------
extern "C" void kernel_launch(void* const* d_in, const int* in_sizes, int n_in,
                              void* d_out, int out_size, void* d_ws, size_t ws_size,
                              hipStream_t stream) {
  const float* input  = (const float*)d_in[0];   // [1024,128]
  const float* adj    = (const float*)d_in[1];   // [1024,1024]
  // d_in[2] = feat_adj: unused by the reference
  const float* weight = (const float*)d_in[3];   // [128,128]
  const float* bias   = (const float*)d_in[4];   // [128]
  const float* w1     = (const float*)d_in[5];   // [128,256]
  const float* b1     = (const float*)d_in[6];   // [128]
  float* out = (float*)d_out;                    // [1024,128]

  // Workspace layout (floats): a, c, S, support  -> ~5.8 MB total
  float* a       = (float*)d_ws;
  float* c       = a + (size_t)NN * FD;
  float* S       = c + (size_t)NN * FD;
  float* support = S + (size_t)NN * NN;

  dim3 g1(128, 2);
  k_ac<<<g1, 128, 0, stream>>>(input, w1, b1, a, c);

  dim3 g2(NN / 32, NN / 32);
  k_sig<<<g2, 256, 0, stream>>>(a, c, adj, S);

  k_support<<<128, 128, 0, stream>>>(S, input, support);

  k_out<<<128, 128, 0, stream>>>(support, weight, bias, out);
}